// Mask_Model_68599217651766
// MI455X (gfx1250) — compile-verified
//
#include <hip/hip_runtime.h>

#define U_NUM 50000
#define I_NUM 50000
#define EMBED 64
#define ROW_TILES (U_NUM / 16)   // 3125, exact

typedef __attribute__((ext_vector_type(2))) float v2f;
typedef __attribute__((ext_vector_type(8))) float v8f;

// ---- monotonic float<->uint mapping for atomic float max via atomicMax(u32) ----
__device__ __forceinline__ unsigned f2ord(float f) {
  unsigned u = __float_as_uint(f);
  return (u & 0x80000000u) ? ~u : (u | 0x80000000u);
}
__device__ __forceinline__ float ord2f(unsigned u) {
  return (u & 0x80000000u) ? __uint_as_float(u & 0x7fffffffu) : __uint_as_float(~u);
}

// ---------------- init segment scratch ----------------
// m_u,m_i are contiguous (2*U_NUM u32), s_u,s_i contiguous (2*U_NUM f32)
__global__ void init_seg(unsigned* __restrict__ m, float* __restrict__ s, int n) {
  int i = blockIdx.x * blockDim.x + threadIdx.x;
  if (i < n) {
    m[i] = f2ord(-INFINITY);
    s[i] = 0.0f;
  }
}

// ---------------- fp32 WMMA projection: O = E @ W^T + b ----------------
// grid.x = row tile (16 rows), grid.y = matrix id:
//   0: Qu = user @ Wq^T + bq     1: Ku = user @ Wk^T + bk
//   2: Qi = item @ Wq^T + bq     3: Ki = item @ Wk^T + bk
// block = 128 (4 waves); wave w computes output columns [16w, 16w+16)
__global__ void proj_wmma(const float* __restrict__ user_e, const float* __restrict__ item_e,
                          const float* __restrict__ Wq, const float* __restrict__ bq,
                          const float* __restrict__ Wk, const float* __restrict__ bk,
                          float* __restrict__ ws) {
  const int mat = blockIdx.y;
  const float* E = (mat < 2) ? user_e : item_e;
  const float* W = (mat & 1) ? Wk : Wq;
  const float* b = (mat & 1) ? bk : bq;
  float* O = ws + (size_t)mat * ((size_t)U_NUM * EMBED);

  const int lane  = threadIdx.x & 31;
  const int n0    = (threadIdx.x >> 5) * 16;   // wave -> output column tile
  const int m0    = blockIdx.x * 16;
  const int l15   = lane & 15;
  const int half  = lane >> 4;                 // 0: K 0..1, 1: K 2..3 per step
  const int khalf = half * 2;

  // C/D layout: lane -> N = n0+l15 (all 8 VGPRs), VGPR j -> M = m0 + j + 8*half.
  // Initialize accumulator with the bias (same column for all 8 elements).
  const float bias = b[n0 + l15];
  v8f c;
#pragma unroll
  for (int j = 0; j < 8; ++j) c[j] = bias;

  // A[16x4]: lane l15 holds row M=m0+l15; VGPR0/1 = E[M, k0+khalf .. +1]
  // B[4x16]: lane l15 holds col N=n0+l15; VGPR0/1 = W^T[k0+khalf, N] = W[N, k0+khalf .. +1]
  const float* arow = E + (size_t)(m0 + l15) * EMBED + khalf;
  const float* brow = W + (size_t)(n0 + l15) * EMBED + khalf;

#pragma unroll
  for (int k0 = 0; k0 < EMBED; k0 += 4) {
    v2f a  = *(const v2f*)(arow + k0);
    v2f bb = *(const v2f*)(brow + k0);
    c = __builtin_amdgcn_wmma_f32_16x16x4_f32(false, a, false, bb, (short)0, c, false, false);
  }

  float* orow = O + (size_t)(m0 + 8 * half) * EMBED + (n0 + l15);
#pragma unroll
  for (int j = 0; j < 8; ++j) orow[(size_t)j * EMBED] = c[j];
}

// ---------------- edge passes ----------------
__device__ __forceinline__ float dot64(const float* __restrict__ a, const float* __restrict__ b) {
  const float4* a4 = (const float4*)a;
  const float4* b4 = (const float4*)b;
  float acc = 0.0f;
#pragma unroll
  for (int i = 0; i < EMBED / 4; ++i) {
    float4 x = a4[i], y = b4[i];
    acc = fmaf(x.x, y.x, acc);
    acc = fmaf(x.y, y.y, acc);
    acc = fmaf(x.z, y.z, acc);
    acc = fmaf(x.w, y.w, acc);
  }
  return acc;
}

__global__ void edge_logits(const float* __restrict__ Qu, const float* __restrict__ Ku,
                            const float* __restrict__ Qi, const float* __restrict__ Ki,
                            const int* __restrict__ rows, const int* __restrict__ cols,
                            const float* __restrict__ u_ui, const float* __restrict__ u_iu,
                            float* __restrict__ out,
                            unsigned* __restrict__ m_u, unsigned* __restrict__ m_i, int nnz) {
  int e = blockIdx.x * blockDim.x + threadIdx.x;
  if (e >= nnz) return;
  const int r = rows[e];
  const int c = cols[e];

  // M_ui: q = user row, k = item col
  float v1 = dot64(Qu + (size_t)r * EMBED, Ki + (size_t)c * EMBED)
             - __logf(-__logf(u_ui[e]));          // TAU == 1
  out[e] = v1;
  atomicMax(m_u + r, f2ord(v1));

  // M_iu: q = item col, k = user row
  float v2 = dot64(Qi + (size_t)c * EMBED, Ku + (size_t)r * EMBED)
             - __logf(-__logf(u_iu[e]));
  out[nnz + e] = v2;
  atomicMax(m_i + c, f2ord(v2));
}

__global__ void edge_exp(const int* __restrict__ rows, const int* __restrict__ cols,
                         float* __restrict__ out,
                         const unsigned* __restrict__ m_u, const unsigned* __restrict__ m_i,
                         float* __restrict__ s_u, float* __restrict__ s_i, int nnz) {
  int e = blockIdx.x * blockDim.x + threadIdx.x;
  if (e >= nnz) return;
  const int r = rows[e];
  const int c = cols[e];

  float e1 = __expf(out[e] - ord2f(m_u[r]));
  out[e] = e1;
  __hip_atomic_fetch_add(s_u + r, e1, __ATOMIC_RELAXED, __HIP_MEMORY_SCOPE_AGENT);

  float e2 = __expf(out[nnz + e] - ord2f(m_i[c]));
  out[nnz + e] = e2;
  __hip_atomic_fetch_add(s_i + c, e2, __ATOMIC_RELAXED, __HIP_MEMORY_SCOPE_AGENT);
}

__global__ void edge_norm(const int* __restrict__ rows, const int* __restrict__ cols,
                          float* __restrict__ out,
                          const float* __restrict__ s_u, const float* __restrict__ s_i, int nnz) {
  int e = blockIdx.x * blockDim.x + threadIdx.x;
  if (e >= nnz) return;
  out[e]       = out[e]       / s_u[rows[e]];
  out[nnz + e] = out[nnz + e] / s_i[cols[e]];
}

// ---------------- launch ----------------
extern "C" void kernel_launch(void* const* d_in, const int* in_sizes, int n_in,
                              void* d_out, int out_size, void* d_ws, size_t ws_size,
                              hipStream_t stream) {
  (void)n_in; (void)out_size; (void)ws_size;

  const float* user_e = (const float*)d_in[0];
  const float* item_e = (const float*)d_in[1];
  const float* Wq     = (const float*)d_in[2];
  const float* bq     = (const float*)d_in[3];
  const float* Wk     = (const float*)d_in[4];
  const float* bk     = (const float*)d_in[5];
  const int*   rows   = (const int*)d_in[6];
  const int*   cols   = (const int*)d_in[7];
  const float* u_ui   = (const float*)d_in[8];
  const float* u_iu   = (const float*)d_in[9];
  const int    nnz    = in_sizes[6];

  float* out = (float*)d_out;

  // workspace layout: Qu | Ku | Qi | Ki | m_u | m_i | s_u | s_i
  const size_t PROJ = (size_t)U_NUM * EMBED;
  float*    ws  = (float*)d_ws;
  float*    Qu  = ws;
  float*    Ku  = Qu + PROJ;
  float*    Qi  = Ku + PROJ;
  float*    Ki  = Qi + PROJ;
  unsigned* m_u = (unsigned*)(Ki + PROJ);
  unsigned* m_i = m_u + U_NUM;
  float*    s_u = (float*)(m_i + I_NUM);
  float*    s_i = s_u + U_NUM;

  init_seg<<<(U_NUM + I_NUM + 255) / 256, 256, 0, stream>>>(m_u, s_u, U_NUM + I_NUM);

  proj_wmma<<<dim3(ROW_TILES, 4), 128, 0, stream>>>(user_e, item_e, Wq, bq, Wk, bk, ws);

  const int eb = (nnz + 255) / 256;
  edge_logits<<<eb, 256, 0, stream>>>(Qu, Ku, Qi, Ki, rows, cols, u_ui, u_iu, out, m_u, m_i, nnz);
  edge_exp<<<eb, 256, 0, stream>>>(rows, cols, out, m_u, m_i, s_u, s_i, nnz);
  edge_norm<<<eb, 256, 0, stream>>>(rows, cols, out, s_u, s_i, nnz);
}